// TimesBlock_36309653521023
// MI455X (gfx1250) — compile-verified
//
#include <hip/hip_runtime.h>
#include <hip/hip_bf16.h>
#include <math.h>

// ---------------------------------------------------------------------------
// Problem constants (from reference)
// ---------------------------------------------------------------------------
#define BB    32
#define CC    64
#define TT    512
#define KK    3
#define PMAXC 512
#define MINP  32
#define CMAXC 16          // T / MIN_P
#define NSEQ  (BB * CC)   // 2048
#define NBINS 257         // rfft bins of T=512
#define NBPAD 272         // 17 * 16, bin dim padded for WMMA M tiles
#define EPSF  1.1920929e-07f
#define NEGF  (-3.4028234663852886e+38f)

typedef float v2f __attribute__((ext_vector_type(2)));
typedef float v8f __attribute__((ext_vector_type(8)));
typedef int   v4i __attribute__((ext_vector_type(4)));

__device__ __forceinline__ float gelu_exact(float x) {
    return 0.5f * x * (1.0f + erff(x * 0.70710678118654752440f));
}

// ---------------------------------------------------------------------------
// gfx1250 async global->LDS path (ASYNCcnt).  Guarded so the file compiles on
// toolchains that lack the builtins; fallback is plain VGPR-staged copies.
// Builtin signature (per compiler diagnostic): (v4i AS1* src, v4i AS3* dst,
// imm offset, imm cpol).
// ---------------------------------------------------------------------------
#if defined(__has_builtin)
#if __has_builtin(__builtin_amdgcn_global_load_async_to_lds_b128) && \
    __has_builtin(__builtin_amdgcn_s_wait_asynccnt)
#define HAS_ASYNC_LDS 1
#endif
#endif
#ifndef HAS_ASYNC_LDS
#define HAS_ASYNC_LDS 0
#endif

__device__ __forceinline__ void async_copy16(const float* gsrc, float* ldst) {
#if HAS_ASYNC_LDS
    __builtin_amdgcn_global_load_async_to_lds_b128(
        (__attribute__((address_space(1))) v4i*)gsrc,
        (__attribute__((address_space(3))) v4i*)ldst,
        0, 0);
#else
    *(float4*)ldst = *(const float4*)gsrc;
#endif
}

#if HAS_ASYNC_LDS
#define ASYNC_WAIT(n) __builtin_amdgcn_s_wait_asynccnt(n)
#else
#define ASYNC_WAIT(n) ((void)0)
#endif

// ---------------------------------------------------------------------------
// Kernel 1: DFT basis tables.  cosT[k][t] = cos(2*pi*k*t/T), sinT = -sin(...)
// Exact argument reduction via (k*t mod 512).  Rows k==0 and k>=257 are zero
// so that amp[0]==0 (matches .at[...,0].set(0)) and padded bins never win topk.
// ---------------------------------------------------------------------------
__global__ void tb_basis_kernel(float* __restrict__ cosT, float* __restrict__ sinT) {
    int i = blockIdx.x * blockDim.x + threadIdx.x;
    if (i >= NBPAD * TT) return;
    int k = i >> 9;
    int t = i & (TT - 1);
    if (k == 0 || k >= NBINS) {
        cosT[i] = 0.0f;
        sinT[i] = 0.0f;
    } else {
        int m = (k * t) & (TT - 1);
        float ang = (float)m * (6.28318530717958647692f / (float)TT);
        cosT[i] = cosf(ang);
        sinT[i] = -sinf(ang);
    }
}

// ---------------------------------------------------------------------------
// Kernel 2: DFT as GEMM on the matrix pipe, async-double-buffered.
//   Re = Basis_cos(272x512) x X^T(512x2048),  Im = Basis_sin x X^T
//   amp[seq][bin] = sqrt(Re^2 + Im^2)
// Block = 256 threads (8 wave32), tile = 16 bins x 128 seqs; each wave owns a
// 16x16 (bin x seq) tile, V_WMMA_F32_16X16X4_F32 over K=512 in chunks of 32.
// Chunks are ping-pong staged with GLOBAL_LOAD_ASYNC_TO_LDS_B128: each thread
// issues exactly 5 b128 loads per chunk, so s_wait_asynccnt 5 (in-order
// completion) guarantees the current buffer while the next stays in flight.
// Row stride 36 floats (144 B): keeps b128 LDS stores 16B-aligned and makes
// both the A reads (stride 36 words mod 64 banks) and B reads conflict-free.
// ---------------------------------------------------------------------------
#define XS 36
__global__ void tb_dft_wmma_kernel(const float* __restrict__ feat,
                                   const float* __restrict__ cosT,
                                   const float* __restrict__ sinT,
                                   float* __restrict__ amp) {
    __shared__ float xlds[2][128 * XS];  // [buf][seq][k], 36-float rows
    __shared__ float clds[2][16 * XS];
    __shared__ float slds[2][16 * XS];

    const int tid  = threadIdx.x;
    const int wv   = tid >> 5;
    const int lane = tid & 31;
    const int bt   = blockIdx.x;          // bin tile   0..16
    const int sb   = blockIdx.y;          // seq block  0..15

    const int mA = lane & 15;             // A-fragment row (bin within tile)
    const int hi = lane >> 4;             // lane half selects K pair
    const int sB = wv * 16 + (lane & 15); // B-fragment column (seq within block)

    v8f accRe = {};
    v8f accIm = {};

    auto issue = [&](int buf, int k0) {
        // X chunk: 128 seqs x 32 floats = 1024 b128 transfers (4 per thread)
        for (int i = tid; i < 1024; i += 256) {
            int s = i >> 3;
            int q = (i & 7) << 2;
            async_copy16(&feat[(size_t)(sb * 128 + s) * TT + k0 + q],
                         &xlds[buf][s * XS + q]);
        }
        // basis chunk: 2 tables x 16 rows x 8 vec4 = 256 (1 per thread)
        {
            int tab = tid >> 7;
            int r   = (tid >> 3) & 15;
            int q   = (tid & 7) << 2;
            const float* src = tab ? sinT : cosT;
            float*       dst = tab ? slds[buf] : clds[buf];
            async_copy16(&src[(size_t)(bt * 16 + r) * TT + k0 + q],
                         &dst[r * XS + q]);
        }
    };

    issue(0, 0);
    issue(1, 32);

    for (int chunk = 0; chunk < 16; ++chunk) {
        const int cur = chunk & 1;
        if (chunk < 15) { ASYNC_WAIT(5); } else { ASYNC_WAIT(0); }
        __syncthreads();

#pragma unroll
        for (int kc = 0; kc < 32; kc += 4) {
            const int kA = kc + 2 * hi;
            v2f ac, as2, bx;
            ac[0]  = clds[cur][mA * XS + kA];
            ac[1]  = clds[cur][mA * XS + kA + 1];
            as2[0] = slds[cur][mA * XS + kA];
            as2[1] = slds[cur][mA * XS + kA + 1];
            bx[0]  = xlds[cur][sB * XS + kA];
            bx[1]  = xlds[cur][sB * XS + kA + 1];
            accRe = __builtin_amdgcn_wmma_f32_16x16x4_f32(false, ac, false, bx,
                                                          (short)0, accRe, false, false);
            accIm = __builtin_amdgcn_wmma_f32_16x16x4_f32(false, as2, false, bx,
                                                          (short)0, accIm, false, false);
        }
        __syncthreads();
        if (chunk + 2 < 16) issue(cur, (chunk + 2) * 32);
    }

    // C layout: VGPR r -> bin = bt*16 + r + 8*hi ; lane&15 -> seq column.
    const int seq     = sb * 128 + wv * 16 + (lane & 15);
    const int binbase = bt * 16 + 8 * hi;
#pragma unroll
    for (int r = 0; r < 8; ++r) {
        float re = accRe[r];
        float im = accIm[r];
        amp[(size_t)seq * NBPAD + binbase + r] = sqrtf(re * re + im * im);
    }
}

// ---------------------------------------------------------------------------
// Kernel 3: fused per-sequence pipeline.  One 256-thread workgroup per (b,c)
// sequence.  The three 16x512 folded period tiles (96 KB fp32) live entirely
// in LDS (320 KB/WGP on gfx1250) so the 201 MB folded tensor never touches
// HBM.  Does: top-3 (tie-break = lowest index, as lax.top_k), period fold,
// 1x1/3x3/5x5 convs + proj + exact GELU + residual, masked avg/max pooling
// over cycles, softmax period reconstruction.  Emits `combined` (B,192,T).
// Sequence + amplitude row are staged via the async global->LDS path.
// ---------------------------------------------------------------------------
// Dynamic LDS layout (floats):
#define L_SEQ   0                  // 512
#define L_FOLD  512                // 3*16*512 = 24576
#define L_AMP   (L_FOLD + 24576)   // 272
#define L_REDV  (L_AMP + 272)      // 256
#define L_REDI  (L_REDV + 256)     // 256 (ints)
#define L_BC    (L_REDI + 256)     // 16  (kidx[3] as float bits, kval[3])
#define L_WB    (L_BC + 16)        // 128 (packed conv weights)
#define L_TOT   (L_WB + 128)       // 26016 floats = 104064 bytes

__device__ __forceinline__ float fold_at(const float* __restrict__ fold,
                                         int k, int h, int w) {
    if ((unsigned)h < (unsigned)CMAXC && (unsigned)w < (unsigned)PMAXC)
        return fold[(k * CMAXC + h) * PMAXC + w];
    return 0.0f;
}

__global__ void tb_fold_conv_pool_kernel(
    const float* __restrict__ feat, const float* __restrict__ amp,
    const float* __restrict__ wp0, const float* __restrict__ bp0,
    const float* __restrict__ wp1, const float* __restrict__ bp1,
    const float* __restrict__ wp2, const float* __restrict__ bp2,
    const float* __restrict__ wpr, const float* __restrict__ bpr,
    const float* __restrict__ wrs, const float* __restrict__ brs,
    float* __restrict__ comb) {
    extern __shared__ float smem[];
    float* seqL = smem + L_SEQ;
    float* fold = smem + L_FOLD;
    float* ampL = smem + L_AMP;
    float* redv = smem + L_REDV;
    int*   redi = (int*)(smem + L_REDI);
    float* bc   = smem + L_BC;
    float* wb   = smem + L_WB;

    const int tid = threadIdx.x;
    const int s   = blockIdx.x;          // sequence id = b*64 + c

    // ---- stage sequence + amplitude row via async LDS path --------------
    if (tid < 128) {
        async_copy16(&feat[(size_t)s * TT + tid * 4], &seqL[tid * 4]);
    } else if (tid < 128 + 68) {
        int j = tid - 128;
        async_copy16(&amp[(size_t)s * NBPAD + j * 4], &ampL[j * 4]);
    }
    // ---- stage packed conv weights --------------------------------------
    if (tid < 3)   wb[tid]       = wp0[tid];
    if (tid == 0)  wb[3]         = bp0[0];
    if (tid < 27)  wb[4 + tid]   = wp1[tid];
    if (tid == 1)  wb[31]        = bp1[0];
    if (tid < 75)  wb[32 + tid]  = wp2[tid];
    if (tid == 2)  wb[107]       = bp2[0];
    if (tid < 3)   wb[108 + tid] = wpr[tid];
    if (tid == 4)  wb[111]       = bpr[0];
    if (tid < 3)   wb[112 + tid] = wrs[tid];
    if (tid == 5)  wb[115]       = brs[0];
    ASYNC_WAIT(0);
    __syncthreads();

    // ---- top-3 amplitudes over bins [0,257); ties -> lowest index -------
    for (int kk = 0; kk < KK; ++kk) {
        float bv = NEGF;
        int   bi = NBINS;
        for (int j = tid; j < NBINS; j += 256) {
            float v = ampL[j];
            if (v > bv || (v == bv && j < bi)) { bv = v; bi = j; }
        }
        redv[tid] = bv;
        redi[tid] = bi;
        __syncthreads();
        for (int st = 128; st > 0; st >>= 1) {
            if (tid < st) {
                float ov = redv[tid + st];
                int   oi = redi[tid + st];
                if (ov > redv[tid] || (ov == redv[tid] && oi < redi[tid])) {
                    redv[tid] = ov; redi[tid] = oi;
                }
            }
            __syncthreads();
        }
        if (tid == 0) {
            bc[kk]     = __int_as_float(redi[0]);  // index
            bc[4 + kk] = redv[0];                  // kamp value
            ampL[redi[0]] = NEGF;                  // exclude from next pass
        }
        __syncthreads();
    }

    // ---- per-branch period parameters ----------------------------------
    int   P[KK], cyc[KK], base[KK];
    float kv[KK];
    for (int k = 0; k < KK; ++k) {
        int idx = __float_as_int(bc[k]);
        if (idx < 1) idx = 1;
        int p = TT / idx;
        p = min(max(p, MINP), PMAXC);
        P[k]    = p;
        cyc[k]  = max(TT / p, 1);
        base[k] = TT - cyc[k] * P[k];   // >= 0
        kv[k]   = bc[4 + k];
    }

    // ---- build folded tiles in LDS (mask applied: invalid -> 0) ---------
    for (int k = 0; k < KK; ++k) {
        for (int i = tid; i < CMAXC * PMAXC; i += 256) {
            int h = i >> 9;
            int w = i & (PMAXC - 1);
            float v = 0.0f;
            if (h < cyc[k] && w < P[k]) {
                int g = base[k] + h * P[k] + w;
                if (g > TT - 1) g = TT - 1;
                v = seqL[g];
            }
            fold[(k * CMAXC + h) * PMAXC + w] = v;
        }
    }
    __syncthreads();

    // ---- softmax over branch amplitudes (branches always valid) ---------
    float mx = fmaxf(kv[0], fmaxf(kv[1], kv[2]));
    float e0 = expf(kv[0] - mx), e1 = expf(kv[1] - mx), e2 = expf(kv[2] - mx);
    float esum = e0 + e1 + e2;
    float bwk[KK] = {e0 / esum, e1 / esum, e2 / esum};

    const int b = s >> 6;
    const int c = s & 63;

    // ---- convs + GELU + residual + pooling; 2 columns per thread --------
    for (int cc = 0; cc < 2; ++cc) {
        const int w = tid + cc * 256;
        float sum_cm = 0.0f, sum_z = 0.0f, maxv = NEGF;
        bool  any = false;
        float bsum[KK] = {0.0f, 0.0f, 0.0f};

        for (int h = 0; h < CMAXC; ++h) {
            float f0 = fold[(0 * CMAXC + h) * PMAXC + w];
            float f1 = fold[(1 * CMAXC + h) * PMAXC + w];
            float f2 = fold[(2 * CMAXC + h) * PMAXC + w];
            bsum[0] += f0; bsum[1] += f1; bsum[2] += f2;

            float p0v = wb[3] + wb[0] * f0 + wb[1] * f1 + wb[2] * f2;

            float p1v = wb[31];
            for (int k = 0; k < KK; ++k)
                for (int dh = -1; dh <= 1; ++dh)
                    for (int dw = -1; dw <= 1; ++dw)
                        p1v += wb[4 + k * 9 + (dh + 1) * 3 + (dw + 1)] *
                               fold_at(fold, k, h + dh, w + dw);

            float p2v = wb[107];
            for (int k = 0; k < KK; ++k)
                for (int dh = -2; dh <= 2; ++dh)
                    for (int dw = -2; dw <= 2; ++dw)
                        p2v += wb[32 + k * 25 + (dh + 2) * 5 + (dw + 2)] *
                               fold_at(fold, k, h + dh, w + dw);

            float z = gelu_exact(wb[108] * p0v + wb[109] * p1v + wb[110] * p2v + wb[111]);
            float rsv = wb[112] * f0 + wb[113] * f1 + wb[114] * f2 + wb[115];
            float z2 = z + rsv;

            bool m0 = (h < cyc[0]) && (w < P[0]);
            bool m1 = (h < cyc[1]) && (w < P[1]);
            bool m2 = (h < cyc[2]) && (w < P[2]);
            float cm = (m0 || m1 || m2) ? 1.0f : 0.0f;
            sum_cm += cm;
            sum_z  += z2 * cm;
            if (cm > 0.0f) { any = true; maxv = fmaxf(maxv, z2); }
        }

        float avgp = sum_z / (sum_cm + EPSF);
        float maxp = any ? maxv : 0.0f;

        float recon = 0.0f;
        for (int k = 0; k < KK; ++k) {
            float bwt = (w < P[k]) ? (float)cyc[k] : 0.0f;
            float br  = (bwt > 0.0f) ? bsum[k] / (bwt + EPSF) : 0.0f;
            recon += br * bwk[k];
        }

        comb[((size_t)b * 192 + c) * TT + w]        = avgp;
        comb[((size_t)b * 192 + 64 + c) * TT + w]   = maxp;
        comb[((size_t)b * 192 + 128 + c) * TT + w]  = recon;
    }
}

// ---------------------------------------------------------------------------
// Kernel 4: fuse GEMM  out[b,o,t] = gelu( W(64x192) @ combined(b,192,512) + bias )
// One wave32 per 16x16 (o x t) tile via V_WMMA_F32_16X16X4_F32, K = 192.
// ---------------------------------------------------------------------------
__global__ void tb_fuse_wmma_kernel(const float* __restrict__ comb,
                                    const float* __restrict__ wfuse,
                                    const float* __restrict__ bfuse,
                                    float* __restrict__ out) {
    const int tid  = threadIdx.x;
    const int wv   = tid >> 5;
    const int lane = tid & 31;
    const int tile = blockIdx.x * 8 + wv;     // 4096 tiles total
    const int bIdx = tile >> 7;               // / 128
    const int rem  = tile & 127;
    const int mt   = rem >> 5;                // o tile 0..3
    const int nt   = rem & 31;                // t tile 0..31

    const int hi = lane >> 4;
    const int mg = mt * 16 + (lane & 15);     // global output channel (A row)
    const int tg = nt * 16 + (lane & 15);     // global t (B column)

    // warm L2/L0 for the operands we are about to stream
    __builtin_prefetch(&wfuse[(size_t)mg * (3 * CC)], 0, 3);
    __builtin_prefetch(&comb[((size_t)bIdx * 192) * TT + tg], 0, 3);

    v8f acc = {};
#pragma unroll 4
    for (int kk = 0; kk < 3 * CC; kk += 4) {
        const int kA = kk + 2 * hi;
        v2f a, bfr;
        a[0]   = wfuse[(size_t)mg * (3 * CC) + kA];
        a[1]   = wfuse[(size_t)mg * (3 * CC) + kA + 1];
        bfr[0] = comb[((size_t)bIdx * 192 + kA) * TT + tg];
        bfr[1] = comb[((size_t)bIdx * 192 + kA + 1) * TT + tg];
        acc = __builtin_amdgcn_wmma_f32_16x16x4_f32(false, a, false, bfr,
                                                    (short)0, acc, false, false);
    }

#pragma unroll
    for (int r = 0; r < 8; ++r) {
        int m = mt * 16 + r + 8 * hi;
        float v = acc[r] + bfuse[m];
        out[((size_t)bIdx * CC + m) * TT + tg] = gelu_exact(v);
    }
}

// ---------------------------------------------------------------------------
// Launch
// ---------------------------------------------------------------------------
extern "C" void kernel_launch(void* const* d_in, const int* in_sizes, int n_in,
                              void* d_out, int out_size, void* d_ws, size_t ws_size,
                              hipStream_t stream) {
    (void)in_sizes; (void)n_in; (void)out_size; (void)ws_size;

    const float* feat = (const float*)d_in[0];
    const float* wp0  = (const float*)d_in[1];
    const float* bp0  = (const float*)d_in[2];
    const float* wp1  = (const float*)d_in[3];
    const float* bp1  = (const float*)d_in[4];
    const float* wp2  = (const float*)d_in[5];
    const float* bp2  = (const float*)d_in[6];
    const float* wpr  = (const float*)d_in[7];
    const float* bpr  = (const float*)d_in[8];
    const float* wrs  = (const float*)d_in[9];
    const float* brs  = (const float*)d_in[10];
    const float* wfu  = (const float*)d_in[11];
    const float* bfu  = (const float*)d_in[12];
    float* out = (float*)d_out;

    float* ws   = (float*)d_ws;
    float* cosT = ws;                                  // 272*512
    float* sinT = cosT + (size_t)NBPAD * TT;           // 272*512
    float* amp  = sinT + (size_t)NBPAD * TT;           // 2048*272
    float* comb = amp  + (size_t)NSEQ * NBPAD;         // 32*192*512

    // 1) DFT basis tables
    tb_basis_kernel<<<(NBPAD * TT + 255) / 256, 256, 0, stream>>>(cosT, sinT);

    // 2) DFT amplitudes via f32 WMMA GEMM (async double-buffered staging)
    dim3 g2(NBPAD / 16, NSEQ / 128);
    tb_dft_wmma_kernel<<<g2, 256, 0, stream>>>(feat, cosT, sinT, amp);

    // 3) fused fold + conv + pool per sequence (104 KB dynamic LDS,
    //    well under gfx1250's 320 KB/WGP)
    tb_fold_conv_pool_kernel<<<NSEQ, 256, L_TOT * sizeof(float), stream>>>(
        feat, amp, wp0, bp0, wp1, bp1, wp2, bp2, wpr, bpr, wrs, brs, comb);

    // 4) fuse GEMM + GELU via f32 WMMA
    tb_fuse_wmma_kernel<<<(BB * 4 * 32) / 8, 256, 0, stream>>>(comb, wfu, bfu, out);
}